// GNN_29970281791606
// MI455X (gfx1250) — compile-verified
//
#include <hip/hip_runtime.h>
#include <hip/hip_bf16.h>

#define N_NODES 100000
#define CDIM 256

typedef __attribute__((ext_vector_type(2))) float v2f;
typedef __attribute__((ext_vector_type(8))) float v8f;

// ---------------- degree / normalization ----------------

__global__ __launch_bounds__(256) void k_init_deg(float* __restrict__ deg) {
    int i = blockIdx.x * 256 + threadIdx.x;
    if (i < N_NODES) deg[i] = 1.0f;  // self-loop contributes 1
}

__global__ __launch_bounds__(256) void k_count_deg(const long long* __restrict__ dst,
                                                   float* __restrict__ deg, long long E) {
    long long e = (long long)blockIdx.x * 256 + threadIdx.x;
    if (e < E) atomicAdd(&deg[dst[e]], 1.0f);
}

__global__ __launch_bounds__(256) void k_rsqrt(float* __restrict__ deg) {
    int i = blockIdx.x * 256 + threadIdx.x;
    if (i < N_NODES) deg[i] = rsqrtf(deg[i]);
}

// ---------------- WMMA f32 GEMM: H[M,256] = X[M,256] @ W[256,256] ----------------
// One wave computes a 16-row x 64-col tile (4 accumulators share the A fragment).
// grid = (M/16, 256/64), block = 32 (one wave, EXEC all-1s as WMMA requires).

__global__ __launch_bounds__(32) void k_gemm(const float* __restrict__ X,
                                             const float* __restrict__ W,
                                             float* __restrict__ H) {
    const int lane = threadIdx.x & 31;
    const int half = lane >> 4;      // 0 = lanes 0-15, 1 = lanes 16-31
    const int mn   = lane & 15;      // M (for A) / N (for B) index within tile
    const long long rowBase = (long long)blockIdx.x * 16;
    const int colBase = blockIdx.y * 64;

    v8f acc0 = {}, acc1 = {}, acc2 = {}, acc3 = {};
    const float* xrow = X + (rowBase + mn) * CDIM;

#pragma unroll 4
    for (int k0 = 0; k0 < CDIM; k0 += 4) {
        const int kA = k0 + 2 * half;
        // A 16x4 f32: lane holds (K=kA, K=kA+1) for row M=mn  -> one 8B load
        float2 av = *(const float2*)(xrow + kA);
        v2f a; a.x = av.x; a.y = av.y;
        // B 4x16 f32: lane holds (K=kA, K=kA+1) for col N, per 16-col tile
        const float* w0 = W + (size_t)kA * CDIM + colBase + mn;
        const float* w1 = w0 + CDIM;
        v2f b0, b1, b2, b3;
        b0.x = w0[0];  b0.y = w1[0];
        b1.x = w0[16]; b1.y = w1[16];
        b2.x = w0[32]; b2.y = w1[32];
        b3.x = w0[48]; b3.y = w1[48];
        acc0 = __builtin_amdgcn_wmma_f32_16x16x4_f32(false, a, false, b0, (short)0, acc0, false, false);
        acc1 = __builtin_amdgcn_wmma_f32_16x16x4_f32(false, a, false, b1, (short)0, acc1, false, false);
        acc2 = __builtin_amdgcn_wmma_f32_16x16x4_f32(false, a, false, b2, (short)0, acc2, false, false);
        acc3 = __builtin_amdgcn_wmma_f32_16x16x4_f32(false, a, false, b3, (short)0, acc3, false, false);
    }

    // C/D layout: VGPR j holds row M = j + 8*half, N = mn (per 16-col tile)
    float* hb = H + rowBase * CDIM + colBase;
#pragma unroll
    for (int j = 0; j < 8; ++j) {
        long long r = j + 8 * half;
        float* hr = hb + r * CDIM + mn;
        hr[0]  = acc0[j];
        hr[16] = acc1[j];
        hr[32] = acc2[j];
        hr[48] = acc3[j];
    }
}

// ---------------- aggregation ----------------

// A[i,:] = H[i,:] * dis[i]^2   (self-loop term, also zero-initializes A)
__global__ __launch_bounds__(256) void k_selfloop(const float* __restrict__ H,
                                                  const float* __restrict__ dis,
                                                  float* __restrict__ A) {
    long long idx = (long long)blockIdx.x * 256 + threadIdx.x;  // float4 index
    if (idx >= (long long)N_NODES * (CDIM / 4)) return;
    long long row = idx >> 6;  // 64 float4 per row
    float s = dis[row];
    s = s * s;
    float4 v = ((const float4*)H)[idx];
    v.x *= s; v.y *= s; v.z *= s; v.w *= s;
    ((float4*)A)[idx] = v;
}

// A[dst,:] += H[src,:] * dis[src]*dis[dst]   (64 threads per edge, float4 lanes)
__global__ __launch_bounds__(256) void k_scatter(const float* __restrict__ H,
                                                 const long long* __restrict__ src,
                                                 const long long* __restrict__ dst,
                                                 const float* __restrict__ dis,
                                                 float* __restrict__ A, long long E) {
    long long e = (long long)blockIdx.x * 4 + (threadIdx.x >> 6);
    if (e >= E) return;
    int t = threadIdx.x & 63;
    long long s = src[e];
    long long d = dst[e];
    float n = dis[s] * dis[d];
    float4 v = ((const float4*)(H + s * CDIM))[t];
    float* op = A + d * CDIM + (long long)t * 4;
    atomicAdd(op + 0, v.x * n);
    atomicAdd(op + 1, v.y * n);
    atomicAdd(op + 2, v.z * n);
    atomicAdd(op + 3, v.w * n);
}

__global__ __launch_bounds__(256) void k_bias_relu(float* __restrict__ A,
                                                   const float* __restrict__ b) {
    long long idx = (long long)blockIdx.x * 256 + threadIdx.x;  // float4 index
    if (idx >= (long long)N_NODES * (CDIM / 4)) return;
    int c4 = (int)(idx & 63);
    float4 bb = ((const float4*)b)[c4];
    float4 v = ((float4*)A)[idx];
    v.x = fmaxf(v.x + bb.x, 0.0f);
    v.y = fmaxf(v.y + bb.y, 0.0f);
    v.z = fmaxf(v.z + bb.z, 0.0f);
    v.w = fmaxf(v.w + bb.w, 0.0f);
    ((float4*)A)[idx] = v;
}

// out[row] = dot(A[row,:], Wl) + bl   (one wave per row)
__global__ __launch_bounds__(32) void k_linear(const float* __restrict__ A,
                                               const float* __restrict__ Wl,
                                               const float* __restrict__ bl,
                                               float* __restrict__ out) {
    int row = blockIdx.x;
    int lane = threadIdx.x;
    const float4* hp = (const float4*)(A + (long long)row * CDIM);
    const float4* wp = (const float4*)Wl;
    float4 h0 = hp[lane],      w0 = wp[lane];
    float4 h1 = hp[lane + 32], w1 = wp[lane + 32];
    float acc = h0.x * w0.x + h0.y * w0.y + h0.z * w0.z + h0.w * w0.w
              + h1.x * w1.x + h1.y * w1.y + h1.z * w1.z + h1.w * w1.w;
#pragma unroll
    for (int off = 16; off > 0; off >>= 1) acc += __shfl_xor(acc, off, 32);
    if (lane == 0) out[row] = acc + bl[0];
}

// ---------------- launch ----------------

extern "C" void kernel_launch(void* const* d_in, const int* in_sizes, int n_in,
                              void* d_out, int out_size, void* d_ws, size_t ws_size,
                              hipStream_t stream) {
    const float* x      = (const float*)d_in[0];
    const long long* ei = (const long long*)d_in[1];   // int64 [2, E]
    const float* W1     = (const float*)d_in[2];
    const float* b1     = (const float*)d_in[3];
    const float* W2     = (const float*)d_in[4];
    const float* b2     = (const float*)d_in[5];
    const float* Wl     = (const float*)d_in[6];
    const float* bl     = (const float*)d_in[7];
    float* out          = (float*)d_out;

    const long long E = (long long)in_sizes[1] / 2;
    const long long* src = ei;
    const long long* dst = ei + E;

    float* H   = (float*)d_ws;                         // [N, C]
    float* A   = H + (size_t)N_NODES * CDIM;           // [N, C]
    float* dis = A + (size_t)N_NODES * CDIM;           // [N]

    const int nBlkN   = (N_NODES + 255) / 256;
    const int nBlkE   = (int)((E + 255) / 256);
    const int nBlkNC4 = (int)(((long long)N_NODES * (CDIM / 4) + 255) / 256);
    const int nBlkEdge = (int)((E + 3) / 4);
    const dim3 gemmGrid(N_NODES / 16, CDIM / 64);

    // normalization
    k_init_deg<<<nBlkN, 256, 0, stream>>>(dis);
    k_count_deg<<<nBlkE, 256, 0, stream>>>(dst, dis, E);
    k_rsqrt<<<nBlkN, 256, 0, stream>>>(dis);

    // layer 1
    k_gemm<<<gemmGrid, 32, 0, stream>>>(x, W1, H);
    k_selfloop<<<nBlkNC4, 256, 0, stream>>>(H, dis, A);
    k_scatter<<<nBlkEdge, 256, 0, stream>>>(H, src, dst, dis, A, E);
    k_bias_relu<<<nBlkNC4, 256, 0, stream>>>(A, b1);

    // layer 2
    k_gemm<<<gemmGrid, 32, 0, stream>>>(A, W2, H);
    k_selfloop<<<nBlkNC4, 256, 0, stream>>>(H, dis, A);
    k_scatter<<<nBlkEdge, 256, 0, stream>>>(H, src, dst, dis, A, E);
    k_bias_relu<<<nBlkNC4, 256, 0, stream>>>(A, b2);

    // output projection
    k_linear<<<N_NODES, 32, 0, stream>>>(A, Wl, bl, out);
}